// U_unfolding_29549374997214
// MI455X (gfx1250) — compile-verified
//
#include <hip/hip_runtime.h>

typedef float v2f __attribute__((ext_vector_type(2)));
typedef float v8f __attribute__((ext_vector_type(8)));

#define NUSERS 1024
#define LPAD 66   // even (8B-aligned float2 frags), 66 % 64 != 0 (bank stagger)

#if __has_builtin(__builtin_amdgcn_global_load_async_to_lds_b32)
#define HAVE_ASYNC_LDS 1
#endif

typedef __attribute__((address_space(1))) int gas_int;
typedef __attribute__((address_space(3))) int las_int;

__device__ __forceinline__ v8f wmma_f32(v2f a, v2f b, v8f c) {
  // D = A(16x4 f32) * B(4x16 f32) + C(16x16 f32)
  return __builtin_amdgcn_wmma_f32_16x16x4_f32(false, a, false, b,
                                               (short)0, c, false, false);
}

// Async global->LDS copy, tracked by ASYNCcnt (CDNA5 GLOBAL_LOAD_ASYNC_TO_LDS_B32)
__device__ __forceinline__ void async_g2l_b32(const float* g, float* l) {
#ifdef HAVE_ASYNC_LDS
  __builtin_amdgcn_global_load_async_to_lds_b32(
      (gas_int*)g, (las_int*)l, 0, 0);
#else
  *l = *g;   // synchronous fallback
#endif
}

__device__ __forceinline__ void wait_async_done() {
#ifdef HAVE_ASYNC_LDS
#if __has_builtin(__builtin_amdgcn_s_wait_asynccnt)
  __builtin_amdgcn_s_wait_asynccnt(0);
#else
  asm volatile("s_wait_asynccnt 0" ::: "memory");
#endif
#endif
}

__global__ __launch_bounds__(128, 1) void u_unfold_kernel(
    const float* __restrict__ A, const float* __restrict__ H,
    const float* __restrict__ P, const float* __restrict__ X,
    const float* __restrict__ Z, const float* __restrict__ O,
    float* __restrict__ out)
{
  __shared__ float sAr[64][LPAD];   // A -> Ainv2 rows -> T rows
  __shared__ float sAi[64][LPAD];
  __shared__ float sIr[64][LPAD];   // identity -> inv(A)
  __shared__ float sIi[64][LPAD];
  __shared__ float sHr[64][LPAD];   // Hbar (async-filled during inversion)
  __shared__ float sHi[64][LPAD];
  __shared__ float sPr[64][16];
  __shared__ float sPi[64][16];
  __shared__ float colr[64];
  __shared__ float coli[64];

  const int u    = blockIdx.x;
  const int tid  = threadIdx.x;
  const int lane = tid & 31;
  const int wave = tid >> 5;
  const int m0   = wave * 16;

  // ---- kick off async Hbar -> LDS now; it completes under the inversion ----
  for (int t = tid; t < 4096; t += 128) {
    int i = t >> 6, j = t & 63;
    async_g2l_b32(&H[t * NUSERS + u],          &sHr[i][j]);
    async_g2l_b32(&H[(4096 + t) * NUSERS + u], &sHi[i][j]);
  }
  // warm L2 for X/Z (first globals touched after the inversion)
  __builtin_prefetch((const char*)(X + u * 8192) + tid * 256, 0, 1);
  __builtin_prefetch((const char*)(Z + u * 8192) + tid * 256, 0, 1);

  // ---------- load A (user-strided), identity, P (pad N 8->16) ----------
  for (int t = tid; t < 4096; t += 128) {
    int i = t >> 6, j = t & 63;
    sAr[i][j] = A[t * NUSERS + u];
    sAi[i][j] = A[(4096 + t) * NUSERS + u];
    sIr[i][j] = (i == j) ? 1.0f : 0.0f;
    sIi[i][j] = 0.0f;
  }
  for (int t = tid; t < 1024; t += 128) {
    int i = t >> 4, j = t & 15;
    sPr[i][j] = (j < 8) ? P[(i * 8 + j) * NUSERS + u] : 0.0f;
    sPi[i][j] = (j < 8) ? P[(512 + i * 8 + j) * NUSERS + u] : 0.0f;
  }
  __syncthreads();

  // ---------- complex Gauss-Jordan (diag-boosted, no pivoting) ----------
  for (int k = 0; k < 64; ++k) {
    if (tid < 64) { colr[tid] = sAr[tid][k]; coli[tid] = sAi[tid][k]; }
    __syncthreads();
    float pr = colr[k], pi = coli[k];
    float inv_d = 1.0f / (pr * pr + pi * pi);
    float qr = pr * inv_d, qi = -pi * inv_d;
    if (tid < 64) {
      float ar = sAr[k][tid], ai = sAi[k][tid];
      sAr[k][tid] = ar * qr - ai * qi;
      sAi[k][tid] = ar * qi + ai * qr;
    } else {
      int j = tid - 64;
      float ar = sIr[k][j], ai = sIi[k][j];
      sIr[k][j] = ar * qr - ai * qi;
      sIi[k][j] = ar * qi + ai * qr;
    }
    __syncthreads();
    if (tid < 64) {
      int j = tid;
      float rkr = sAr[k][j], rki = sAi[k][j];
      for (int i = 0; i < 64; ++i) {
        if (i == k) continue;
        float fr = colr[i], fi = coli[i];
        sAr[i][j] -= fr * rkr - fi * rki;
        sAi[i][j] -= fr * rki + fi * rkr;
      }
    } else {
      int j = tid - 64;
      float rkr = sIr[k][j], rki = sIi[k][j];
      for (int i = 0; i < 64; ++i) {
        if (i == k) continue;
        float fr = colr[i], fi = coli[i];
        sIr[i][j] -= fr * rkr - fi * rki;
        sIi[i][j] -= fr * rki + fi * rkr;
      }
    }
    __syncthreads();
  }
  // sI* now holds inv(A)

  // Per-lane fragment geometry (f32 WMMA 16x16x4):
  //  A-frag: m = m0 + (lane&15), k pair = kt*4 + (lane>>4)*2
  //  B-frag: n = (lane&15),      k pair = kt*4 + (lane>>4)*2
  //  C/D:    m = v + 8*(lane>>4), n = lane&15
  const int mA     = m0 + (lane & 15);
  const int koff   = (lane >> 4) * 2;
  const int ncol   = lane & 15;
  const int rowsel = (lane >> 4) * 8;

  // ---------- step 3: Ainv2 rows = X_rows @ invA + Z_rows ----------
  v8f accr[4], acci[4];
  const float* Zr = Z + (u * 2 + 0) * 4096;
  const float* Zi = Z + (u * 2 + 1) * 4096;
  #pragma unroll
  for (int nt = 0; nt < 4; ++nt) {
    #pragma unroll
    for (int v = 0; v < 8; ++v) {
      int m = m0 + v + rowsel;
      int n = nt * 16 + ncol;
      accr[nt][v] = Zr[m * 64 + n];
      acci[nt][v] = Zi[m * 64 + n];
    }
  }
  const float* Xr = X + (u * 2 + 0) * 4096 + mA * 64;
  const float* Xi = X + (u * 2 + 1) * 4096 + mA * 64;
  for (int kt = 0; kt < 16; ++kt) {
    int kk = kt * 4 + koff;
    v2f ax, axi, naxi;
    ax.x  = Xr[kk]; ax.y  = Xr[kk + 1];
    axi.x = Xi[kk]; axi.y = Xi[kk + 1];
    naxi = -axi;   // f32 WMMA has no A/B neg modifier -> negate in VALU
    #pragma unroll
    for (int nt = 0; nt < 4; ++nt) {
      int nB = nt * 16 + ncol;
      v2f br, bi;
      br.x = sIr[kk][nB]; br.y = sIr[kk + 1][nB];
      bi.x = sIi[kk][nB]; bi.y = sIi[kk + 1][nB];
      accr[nt] = wmma_f32(ax,   br, accr[nt]);
      accr[nt] = wmma_f32(naxi, bi, accr[nt]);
      acci[nt] = wmma_f32(ax,   bi, acci[nt]);
      acci[nt] = wmma_f32(axi,  br, acci[nt]);
    }
  }

  // park Ainv2 block in own sA rows (wave-local; original A is dead)
  #pragma unroll
  for (int nt = 0; nt < 4; ++nt) {
    #pragma unroll
    for (int v = 0; v < 8; ++v) {
      int m = m0 + v + rowsel;
      int n = nt * 16 + ncol;
      sAr[m][n] = accr[nt][v];
      sAi[m][n] = acci[nt][v];
    }
  }

  // Hbar async loads must be complete and visible to every wave
  wait_async_done();
  __syncthreads();

  // ---------- step 4: T rows = Ainv2_rows @ Hbar ----------
  v8f zero = {0.f, 0.f, 0.f, 0.f, 0.f, 0.f, 0.f, 0.f};
  v8f tr[4], ti[4];
  #pragma unroll
  for (int nt = 0; nt < 4; ++nt) { tr[nt] = zero; ti[nt] = zero; }
  for (int kt = 0; kt < 16; ++kt) {
    int kk = kt * 4 + koff;
    v2f ar2, ai2, nai2;
    ar2.x = sAr[mA][kk]; ar2.y = sAr[mA][kk + 1];
    ai2.x = sAi[mA][kk]; ai2.y = sAi[mA][kk + 1];
    nai2 = -ai2;
    #pragma unroll
    for (int nt = 0; nt < 4; ++nt) {
      int nB = nt * 16 + ncol;
      v2f br, bi;
      br.x = sHr[kk][nB]; br.y = sHr[kk + 1][nB];
      bi.x = sHi[kk][nB]; bi.y = sHi[kk + 1][nB];
      tr[nt] = wmma_f32(ar2,  br, tr[nt]);
      tr[nt] = wmma_f32(nai2, bi, tr[nt]);
      ti[nt] = wmma_f32(ar2,  bi, ti[nt]);
      ti[nt] = wmma_f32(ai2,  br, ti[nt]);
    }
  }
  // park T block in own sA rows (wave-local; LDS is in-order per wave)
  #pragma unroll
  for (int nt = 0; nt < 4; ++nt) {
    #pragma unroll
    for (int v = 0; v < 8; ++v) {
      int m = m0 + v + rowsel;
      int n = nt * 16 + ncol;
      sAr[m][n] = tr[nt][v];
      sAi[m][n] = ti[nt][v];
    }
  }

  // ---------- step 5: U rows = T_rows @ P + O_rows ----------
  v8f ur, ui;
  const float* Or = O + (u * 2 + 0) * 512;
  const float* Oi = O + (u * 2 + 1) * 512;
  #pragma unroll
  for (int v = 0; v < 8; ++v) {
    int m = m0 + v + rowsel;
    ur[v] = (ncol < 8) ? Or[m * 8 + ncol] : 0.0f;
    ui[v] = (ncol < 8) ? Oi[m * 8 + ncol] : 0.0f;
  }
  for (int kt = 0; kt < 16; ++kt) {
    int kk = kt * 4 + koff;
    v2f ar2, ai2, nai2, br, bi;
    ar2.x = sAr[mA][kk]; ar2.y = sAr[mA][kk + 1];
    ai2.x = sAi[mA][kk]; ai2.y = sAi[mA][kk + 1];
    nai2 = -ai2;
    br.x = sPr[kk][ncol]; br.y = sPr[kk + 1][ncol];
    bi.x = sPi[kk][ncol]; bi.y = sPi[kk + 1][ncol];
    ur = wmma_f32(ar2,  br, ur);
    ur = wmma_f32(nai2, bi, ur);
    ui = wmma_f32(ar2,  bi, ui);
    ui = wmma_f32(ai2,  br, ui);
  }
  // out[2][64][8][1024]
  #pragma unroll
  for (int v = 0; v < 8; ++v) {
    int m = m0 + v + rowsel;
    if (ncol < 8) {
      out[(m * 8 + ncol) * NUSERS + u]        = ur[v];
      out[((64 + m) * 8 + ncol) * NUSERS + u] = ui[v];
    }
  }
}

extern "C" void kernel_launch(void* const* d_in, const int* in_sizes, int n_in,
                              void* d_out, int out_size, void* d_ws, size_t ws_size,
                              hipStream_t stream) {
  const float* A = (const float*)d_in[0];   // [2,64,64,1024]
  const float* H = (const float*)d_in[1];   // [2,64,64,1024]
  const float* P = (const float*)d_in[2];   // [2,64,8,1024]
  const float* X = (const float*)d_in[3];   // [1024,2,64,64]
  const float* Z = (const float*)d_in[4];   // [1024,2,64,64]
  const float* O = (const float*)d_in[5];   // [1024,2,64,8]
  float* out = (float*)d_out;               // [2,64,8,1024]
  u_unfold_kernel<<<dim3(NUSERS), dim3(128), 0, stream>>>(A, H, P, X, Z, O, out);
}